// GNN_82532091560013
// MI455X (gfx1250) — compile-verified
//
#include <hip/hip_runtime.h>

#define N_ENT  100000
#define N_RELx2 400
#define DDIM   64
#define NEDGE  1000000
#define NSHORT 4096
#define NBATCH 32

typedef float v2f __attribute__((ext_vector_type(2)));
typedef float v8f __attribute__((ext_vector_type(8)));

// ---------------------------------------------------------------------------
// C[M,N] = A[M,K] @ B[K,N] (+bias)(+relu) via V_WMMA_F32_16X16X4_F32.
// One wave32 computes one 16x16 output tile. M%16==0, N%16==0, K%4==0.
// A-frag (16x4 f32): lane L holds M=L&15, K-pair = k + (L>>4)*2.
// B-frag (4x16 f32): lane L holds N=L&15, same K-pair split across lane halves.
// C/D (16x16 f32):   vgpr r, lane L -> M = r + (L>=16)*8, N = L&15.
// ---------------------------------------------------------------------------
__global__ void k_gemm_wmma(const float* __restrict__ A, const float* __restrict__ B,
                            const float* __restrict__ bias, float* __restrict__ C,
                            int M, int K, int N, int do_relu)
{
    const int lane = threadIdx.x & 31;
    const int wave = threadIdx.x >> 5;
    const int wpb  = blockDim.x >> 5;
    const int tile = blockIdx.x * wpb + wave;
    const int tn   = N >> 4;
    const int m0   = (tile / tn) << 4;
    const int n0   = (tile % tn) << 4;
    if (m0 >= M) return;                      // wave-uniform: EXEC stays all-1s
    const int half = lane >> 4;               // 0 or 1
    const int l15  = lane & 15;
    const float* __restrict__ arow = A + (size_t)(m0 + l15) * K;

    v8f c = {};
    for (int k = 0; k < K; k += 4) {
        const int ka = k + half * 2;
        v2f a, b;
        a.x = arow[ka];
        a.y = arow[ka + 1];
        b.x = B[(size_t)ka * N + n0 + l15];
        b.y = B[(size_t)(ka + 1) * N + n0 + l15];
        c = __builtin_amdgcn_wmma_f32_16x16x4_f32(
                /*neg_a=*/false, a, /*neg_b=*/false, b,
                /*c_mod=*/(short)0, c, /*reuse_a=*/false, /*reuse_b=*/false);
    }
    const float bb = bias ? bias[n0 + l15] : 0.0f;
#pragma unroll
    for (int r = 0; r < 8; ++r) {
        const int m = m0 + r + half * 8;
        float v = c[r] + bb;
        if (do_relu) v = fmaxf(v, 0.0f);
        C[(size_t)m * N + n0 + l15] = v;
    }
}

// acc[n][d] = bias[d]
__global__ void k_fill_bias(float* __restrict__ acc, const float* __restrict__ bias, int total)
{
    int i = blockIdx.x * blockDim.x + threadIdx.x;
    if (i < total) acc[i] = bias[i & (DDIM - 1)];
}

// for each edge e: acc[dst[e]][:] += y[src[e]][:]  (16 threads/edge, float4 each)
__global__ void k_scatter(const float* __restrict__ y, const int* __restrict__ src,
                          const int* __restrict__ dst, float* __restrict__ acc, int nE)
{
    int i = blockIdx.x * blockDim.x + threadIdx.x;
    if (i >= nE * 16) return;
    const int e = i >> 4;
    const int j = (i & 15) << 2;
    const int s = src[e];
    const int t = dst[e];
    const float4 v = *(const float4*)(y + (size_t)s * DDIM + j);
    float* a = acc + (size_t)t * DDIM + j;
    atomicAdd(a + 0, v.x);
    atomicAdd(a + 1, v.y);
    atomicAdd(a + 2, v.z);
    atomicAdd(a + 3, v.w);
}

__global__ void k_relu(const float* __restrict__ acc, float* __restrict__ x, int total)
{
    int i = blockIdx.x * blockDim.x + threadIdx.x;
    if (i < total) x[i] = fmaxf(acc[i], 0.0f);
}

// h_mm[row] = concat(x[src[si]], rel[etype[si]], x[dst[si]])  (48 float4 per row)
__global__ void k_build_hmm(const float* __restrict__ x, const float* __restrict__ rel,
                            const int* __restrict__ src, const int* __restrict__ dst,
                            const int* __restrict__ etype, const int* __restrict__ sidx,
                            float* __restrict__ h)
{
    int i = blockIdx.x * blockDim.x + threadIdx.x;
    if (i >= NSHORT * 48) return;
    const int row = i / 48;
    const int col = (i % 48) << 2;
    const int e   = sidx[row];
    const float* p;
    if (col < DDIM)          p = x   + (size_t)src[e]   * DDIM + col;
    else if (col < 2 * DDIM) p = rel + (size_t)etype[e] * DDIM + (col - DDIM);
    else                     p = x   + (size_t)dst[e]   * DDIM + (col - 2 * DDIM);
    *(float4*)(h + (size_t)row * (3 * DDIM) + col) = *(const float4*)p;
}

// dueling epilogue for mm head: q = val + adv - mean(adv), adv dim 3
__global__ void k_mm_final(const float* __restrict__ h, const float* __restrict__ Wadv,
                           const float* __restrict__ badv, const float* __restrict__ Wval,
                           const float* __restrict__ bval, float* __restrict__ out)
{
    int row = blockIdx.x * blockDim.x + threadIdx.x;
    if (row >= NSHORT) return;
    const float* hr = h + (size_t)row * DDIM;
    float a0 = badv[0], a1 = badv[1], a2 = badv[2], v = bval[0];
    for (int k = 0; k < DDIM; ++k) {
        const float hv = hr[k];
        a0 = fmaf(hv, Wadv[k * 3 + 0], a0);
        a1 = fmaf(hv, Wadv[k * 3 + 1], a1);
        a2 = fmaf(hv, Wadv[k * 3 + 2], a2);
        v  = fmaf(hv, Wval[k], v);
    }
    const float mean = (a0 + a1 + a2) * (1.0f / 3.0f);
    out[row * 3 + 0] = v + a0 - mean;
    out[row * 3 + 1] = v + a1 - mean;
    out[row * 3 + 2] = v + a2 - mean;
}

// explore head: 32 rows, full dueling MLP per thread (tiny)
__global__ void k_ex_head(const float* __restrict__ x, const int* __restrict__ agent,
                          const float* __restrict__ W0, const float* __restrict__ b0,
                          const float* __restrict__ W1, const float* __restrict__ b1,
                          const float* __restrict__ Wadv, const float* __restrict__ badv,
                          const float* __restrict__ Wval, const float* __restrict__ bval,
                          float* __restrict__ out)
{
    int i = blockIdx.x * blockDim.x + threadIdx.x;
    if (i >= NBATCH) return;
    const float* xr = x + (size_t)agent[i] * DDIM;
    float h0[DDIM], h1[DDIM];
    for (int j = 0; j < DDIM; ++j) {
        float s = b0[j];
        for (int k = 0; k < DDIM; ++k) s = fmaf(xr[k], W0[k * DDIM + j], s);
        h0[j] = fmaxf(s, 0.0f);
    }
    for (int j = 0; j < DDIM; ++j) {
        float s = b1[j];
        for (int k = 0; k < DDIM; ++k) s = fmaf(h0[k], W1[k * DDIM + j], s);
        h1[j] = fmaxf(s, 0.0f);
    }
    float adv[5], mean = 0.0f;
    for (int j = 0; j < 5; ++j) {
        float s = badv[j];
        for (int k = 0; k < DDIM; ++k) s = fmaf(h1[k], Wadv[k * 5 + j], s);
        adv[j] = s;
        mean += s;
    }
    mean *= 0.2f;
    float v = bval[0];
    for (int k = 0; k < DDIM; ++k) v = fmaf(h1[k], Wval[k], v);
    for (int j = 0; j < 5; ++j) out[i * 5 + j] = v + adv[j] - mean;
}

extern "C" void kernel_launch(void* const* d_in, const int* in_sizes, int n_in,
                              void* d_out, int out_size, void* d_ws, size_t ws_size,
                              hipStream_t stream)
{
    (void)in_sizes; (void)n_in; (void)out_size; (void)ws_size;

    const float* ent    = (const float*)d_in[0];
    const float* rel    = (const float*)d_in[1];
    const float* gW     = (const float*)d_in[2];   // [2,64,64]
    const float* gb     = (const float*)d_in[3];   // [2,64]
    const float* mmW0   = (const float*)d_in[4];
    const float* mmb0   = (const float*)d_in[5];
    const float* mmW1   = (const float*)d_in[6];
    const float* mmb1   = (const float*)d_in[7];
    const float* mmWadv = (const float*)d_in[8];
    const float* mmbadv = (const float*)d_in[9];
    const float* mmWval = (const float*)d_in[10];
    const float* mmbval = (const float*)d_in[11];
    const float* exW0   = (const float*)d_in[12];
    const float* exb0   = (const float*)d_in[13];
    const float* exW1   = (const float*)d_in[14];
    const float* exb1   = (const float*)d_in[15];
    const float* exWadv = (const float*)d_in[16];
    const float* exbadv = (const float*)d_in[17];
    const float* exWval = (const float*)d_in[18];
    const float* exbval = (const float*)d_in[19];
    const int*   edge   = (const int*)d_in[20];    // [2,E]
    const int*   etyp   = (const int*)d_in[21];
    const int*   sidx   = (const int*)d_in[22];
    const int*   aidx   = (const int*)d_in[23];
    float*       out    = (float*)d_out;

    const int* src = edge;
    const int* dst = edge + NEDGE;

    // workspace partition (all offsets 256B-aligned)
    float* ws   = (float*)d_ws;
    float* buf0 = ws;                                   // N*64  (y / scratch)
    float* buf1 = buf0 + (size_t)N_ENT * DDIM;          // N*64  (accumulator)
    float* buf2 = buf1 + (size_t)N_ENT * DDIM;          // N*64  (activations x)
    float* hmm  = buf2 + (size_t)N_ENT * DDIM;          // NS*192
    float* h0   = hmm  + (size_t)NSHORT * 3 * DDIM;     // NS*64
    float* h1   = h0   + (size_t)NSHORT * DDIM;         // NS*64

    const int total   = N_ENT * DDIM;                   // 6.4M
    const int gtiles  = (N_ENT / 16) * (DDIM / 16);     // 25000 waves
    const int gblocks = (gtiles + 7) / 8;               // 8 waves / 256-thread block
    const int eblocks = (NEDGE * 16 + 255) / 256;
    const int nblocks = (total + 255) / 256;

    // ---- GCN layer 0: x = relu(scatter(dst, (ent @ W0)[src]) + b0)
    k_gemm_wmma<<<gblocks, 256, 0, stream>>>(ent, gW, nullptr, buf0, N_ENT, DDIM, DDIM, 0);
    k_fill_bias<<<nblocks, 256, 0, stream>>>(buf1, gb, total);
    k_scatter  <<<eblocks, 256, 0, stream>>>(buf0, src, dst, buf1, NEDGE);
    k_relu     <<<nblocks, 256, 0, stream>>>(buf1, buf2, total);

    // ---- GCN layer 1
    k_gemm_wmma<<<gblocks, 256, 0, stream>>>(buf2, gW + DDIM * DDIM, nullptr, buf0, N_ENT, DDIM, DDIM, 0);
    k_fill_bias<<<nblocks, 256, 0, stream>>>(buf1, gb + DDIM, total);
    k_scatter  <<<eblocks, 256, 0, stream>>>(buf0, src, dst, buf1, NEDGE);
    k_relu     <<<nblocks, 256, 0, stream>>>(buf1, buf2, total);          // x2 = buf2

    // ---- memory-management head
    k_build_hmm<<<(NSHORT * 48 + 255) / 256, 256, 0, stream>>>(buf2, rel, src, dst, etyp, sidx, hmm);
    const int t1 = (NSHORT / 16) * (DDIM / 16);         // 1024 tiles
    k_gemm_wmma<<<(t1 + 7) / 8, 256, 0, stream>>>(hmm, mmW0, mmb0, h0, NSHORT, 3 * DDIM, DDIM, 1);
    k_gemm_wmma<<<(t1 + 7) / 8, 256, 0, stream>>>(h0,  mmW1, mmb1, h1, NSHORT, DDIM, DDIM, 1);
    k_mm_final <<<(NSHORT + 255) / 256, 256, 0, stream>>>(h1, mmWadv, mmbadv, mmWval, mmbval, out);

    // ---- explore head (32 rows)
    k_ex_head<<<1, 32, 0, stream>>>(buf2, aidx, exW0, exb0, exW1, exb1,
                                    exWadv, exbadv, exWval, exbval, out + NSHORT * 3);
}